// Attention_6390911337261
// MI455X (gfx1250) — compile-verified
//
#include <hip/hip_runtime.h>
#include <math.h>

// Problem constants (match reference)
#define B_   32
#define SMAX 2048
#define H_   16
#define D_   128
#define E_   2048          // H*D
#define K_   2048          // GEMM reduction dim (= E)

typedef __attribute__((ext_vector_type(2))) float v2f;
typedef __attribute__((ext_vector_type(4))) float v4f;
typedef __attribute__((ext_vector_type(8))) float v8f;

// ---------------------------------------------------------------------------
// GEMM: C[32, N] = A[32, 2048] * W[2048, N] + bias[N]   (fp32 WMMA 16x16x4)
// One wave computes a 16x16 tile; 8 waves per block cover 128 columns.
// grid = (N/128, 32/16), block = 256 threads (8 wave32s).
//
// A-frag (16x4 f32): lanes 0-15 -> rows m0..m0+15 @ {K,K+1}; lanes 16-31 same
// rows @ {K+2,K+3}.  B-frag (4x16): lanes 0-15 -> row K cols n0..n0+15 (VGPR0)
// and row K+1 (VGPR1); lanes 16-31 -> rows K+2/K+3.
// C/D (16x16 f32): VGPR r, lanes 0-15 -> M=r; lanes 16-31 -> M=r+8.
// ---------------------------------------------------------------------------
template <int N_TOTAL>
__global__ __launch_bounds__(256) void gemm_wmma_f32(
    const float* __restrict__ A,     // [32, 2048] row-major
    const float* __restrict__ W,     // [2048, N_TOTAL] row-major
    const float* __restrict__ bias,  // [N_TOTAL]
    float* __restrict__ C)           // [32, N_TOTAL]
{
    const int wave   = threadIdx.x >> 5;
    const int lane   = threadIdx.x & 31;
    const int lane16 = lane & 15;
    const int hi     = lane >> 4;              // 0: K+0/1, 1: K+2/3

    const int n0 = blockIdx.x * 128 + wave * 16;
    const int m0 = blockIdx.y * 16;

    const float* aPtr = A + (size_t)(m0 + lane16) * K_ + 2 * hi;
    const float* bPtr = W + (size_t)(2 * hi) * N_TOTAL + n0 + lane16;

    v8f c = {};
#pragma unroll 4
    for (int k = 0; k < K_; k += 4) {
        v2f a, b;
        // A: two consecutive K elements of this lane's row (merges to b64)
        a.x = aPtr[k];
        a.y = aPtr[k + 1];
        // W is streamed exactly once per M-tile: use non-temporal loads
        b.x = __builtin_nontemporal_load(bPtr + (size_t)k * N_TOTAL);
        b.y = __builtin_nontemporal_load(bPtr + (size_t)(k + 1) * N_TOTAL);
        c = __builtin_amdgcn_wmma_f32_16x16x4_f32(
                /*neg_a=*/false, a, /*neg_b=*/false, b,
                /*c_mod=*/(short)0, c, /*reuse_a=*/false, /*reuse_b=*/false);
    }

    const float bcol = bias[n0 + lane16];
#pragma unroll
    for (int r = 0; r < 8; ++r) {
        const int row = m0 + r + 8 * hi;
        C[(size_t)row * N_TOTAL + n0 + lane16] = c[r] + bcol;
    }
}

// ---------------------------------------------------------------------------
// Decode attention with online softmax.
// grid = B*H blocks, block = 256 threads (8 waves).
// Lane owns a 4-float chunk of D (lane*4): each wave load of K/V is a fully
// coalesced 512B burst. Waves stride the sequence; cache row at pos-1 is
// replaced by the freshly projected k/v (reference's cache update).
// ---------------------------------------------------------------------------
__global__ __launch_bounds__(256) void attn_decode_kernel(
    const float* __restrict__ qkv,      // [B, 3E] from QKV projection
    const float* __restrict__ k_cache,  // [B, S, H, D]
    const float* __restrict__ v_cache,  // [B, S, H, D]
    const int*   __restrict__ input_pos,// [B]
    float* __restrict__ attn_out)       // [B, E]
{
    const int bh   = blockIdx.x;
    const int b    = bh / H_;
    const int h    = bh % H_;
    const int tid  = threadIdx.x;
    const int wave = tid >> 5;
    const int lane = tid & 31;
    const int d0   = lane * 4;

    const int pos = input_pos[b];            // 1..SMAX
    const float scale = 0.088388347648318447f; // 1/sqrt(128)

    const float* qp = qkv + (size_t)b * (3 * E_) + (size_t)h * D_;
    const float* kN = qp + E_;               // fresh k for this (b,h)
    const float* vN = qp + 2 * E_;           // fresh v

    const v4f q4 = *(const v4f*)(qp + d0);

    const size_t rowStride = (size_t)H_ * D_;  // 2048 floats between seq rows
    const float* kBase = k_cache + (size_t)b * SMAX * rowStride + (size_t)h * D_ + d0;
    const float* vBase = v_cache + (size_t)b * SMAX * rowStride + (size_t)h * D_ + d0;

    float m = -INFINITY, l = 0.f;
    v4f acc = {0.f, 0.f, 0.f, 0.f};

    for (int s = wave; s < pos; s += 8) {
        const bool fresh = (s == pos - 1);
        const float* kp = fresh ? (kN + d0) : (kBase + (size_t)s * rowStride);
        const float* vp = fresh ? (vN + d0) : (vBase + (size_t)s * rowStride);

        // streaming reads: non-temporal b128
        const v4f k4 = __builtin_nontemporal_load((const v4f*)kp);
        const v4f v4 = __builtin_nontemporal_load((const v4f*)vp);

        float dot = q4.x * k4.x + q4.y * k4.y + q4.z * k4.z + q4.w * k4.w;
#pragma unroll
        for (int off = 16; off > 0; off >>= 1)    // wave32 butterfly reduce
            dot += __shfl_xor(dot, off, 32);

        const float sc   = dot * scale;
        const float mN   = fmaxf(m, sc);
        const float corr = __expf(m - mN);
        const float p    = __expf(sc - mN);
        acc = acc * corr + p * v4;
        l = l * corr + p;
        m = mN;
    }

    // Cross-wave merge through LDS
    __shared__ float sm[8];
    __shared__ float sl[8];
    __shared__ float sacc[8][D_];

    sacc[wave][d0 + 0] = acc.x;
    sacc[wave][d0 + 1] = acc.y;
    sacc[wave][d0 + 2] = acc.z;
    sacc[wave][d0 + 3] = acc.w;
    if (lane == 0) { sm[wave] = m; sl[wave] = l; }
    __syncthreads();

    if (tid < 32) {
        float M = -INFINITY;
#pragma unroll
        for (int w = 0; w < 8; ++w) M = fmaxf(M, sm[w]);
        float L = 0.f;
        v4f o = {0.f, 0.f, 0.f, 0.f};
#pragma unroll
        for (int w = 0; w < 8; ++w) {
            const float cw = __expf(sm[w] - M);   // 0 for waves with no work
            L   += sl[w] * cw;
            o.x += cw * sacc[w][d0 + 0];
            o.y += cw * sacc[w][d0 + 1];
            o.z += cw * sacc[w][d0 + 2];
            o.w += cw * sacc[w][d0 + 3];
        }
        const float inv = 1.f / L;
        v4f r = o * inv;
        *(v4f*)(attn_out + (size_t)b * E_ + (size_t)h * D_ + d0) = r;
    }
}

// ---------------------------------------------------------------------------
// Launch: QKV projection -> attention -> output projection
// ---------------------------------------------------------------------------
extern "C" void kernel_launch(void* const* d_in, const int* in_sizes, int n_in,
                              void* d_out, int out_size, void* d_ws, size_t ws_size,
                              hipStream_t stream) {
    const float* x         = (const float*)d_in[0];  // [32,1,2048]
    const float* k_cache   = (const float*)d_in[1];  // [32,2048,16,128]
    const float* v_cache   = (const float*)d_in[2];
    const float* W_in      = (const float*)d_in[3];  // [2048,6144]
    const float* b_in      = (const float*)d_in[4];  // [6144]
    const float* W_out     = (const float*)d_in[5];  // [2048,2048]
    const float* b_out     = (const float*)d_in[6];  // [2048]
    const int*   input_pos = (const int*)d_in[7];    // [32]
    float*       out       = (float*)d_out;          // [32,1,2048]

    float* qkv  = (float*)d_ws;                      // [32, 6144]
    float* attn = qkv + (size_t)B_ * 3 * E_;         // [32, 2048]

    // qkv = x @ W_in + b_in
    gemm_wmma_f32<3 * E_><<<dim3((3 * E_) / 128, B_ / 16), 256, 0, stream>>>(
        x, W_in, b_in, qkv);

    // flash attention over KV cache (fresh k/v substituted at pos-1)
    attn_decode_kernel<<<dim3(B_ * H_), 256, 0, stream>>>(
        qkv, k_cache, v_cache, input_pos, attn);

    // out = attn @ W_out + b_out
    gemm_wmma_f32<E_><<<dim3(E_ / 128, B_ / 16), 256, 0, stream>>>(
        attn, W_out, b_out, out);
}